// SparseMemory_84799834293120
// MI455X (gfx1250) — compile-verified
//
#include <hip/hip_runtime.h>
#include <math.h>

typedef __attribute__((ext_vector_type(2))) float v2f;
typedef __attribute__((ext_vector_type(4))) float v4f;
typedef __attribute__((ext_vector_type(8))) float v8f;

#define BATCH 16
#define ISZ   1024
#define MSZ   32768
#define DSZ   128
#define KTOP  32
#define EPSV  1e-8f

// workspace layout (float offsets)
#define WS_WV  0                       // write_vector [16,128]
#define WS_RQ  (WS_WV + BATCH*DSZ)     // read_query   [16,128]
#define WS_QN  (WS_RQ + BATCH*DSZ)     // query norms  [16]
#define WS_MN  (WS_QN + 16)            // memory norms [32768]
#define WS_S   (WS_MN + MSZ)           // write strength [16]
#define WS_SIM (WS_S + 16)             // similarity [16,32768]

// ---------------------------------------------------------------------------
// Kernel 1: fused projections  wv = x@Ww.T+bw, rq = x@Wr.T+br  via WMMA f32
// one wave per 16x16 output tile; 16 waves total (2 proj x 8 N-tiles)
// ---------------------------------------------------------------------------
__global__ __launch_bounds__(512) void k_proj(const float* __restrict__ x,
                                              const float* __restrict__ Ww,
                                              const float* __restrict__ bw,
                                              const float* __restrict__ Wr,
                                              const float* __restrict__ br,
                                              float* __restrict__ ws) {
  const int wave = threadIdx.x >> 5;
  const int lane = threadIdx.x & 31;
  const int proj = wave >> 3;          // 0 = write proj, 1 = read proj
  const int ntile = wave & 7;

  const float* W    = proj ? Wr : Ww;
  const float* bias = proj ? br : bw;
  float* out = ws + (proj ? WS_RQ : WS_WV);

  const int m    = lane & 15;          // A row
  const int koff = (lane >> 4) << 1;   // K sub-offset within 4-wide step
  const int n    = (ntile << 4) + (lane & 15);

  const float* xrow = x + m * ISZ;
  const float* wrow = W + n * ISZ;     // B[k][n] = W[n][k]

  v8f acc = {};
  for (int k0 = 0; k0 < ISZ; k0 += 4) {
    v2f a = *(const v2f*)(xrow + k0 + koff);
    v2f b = *(const v2f*)(wrow + k0 + koff);
    acc = __builtin_amdgcn_wmma_f32_16x16x4_f32(false, a, false, b,
                                                (short)0, acc, false, false);
  }
  const float bv = bias[n];
#pragma unroll
  for (int r = 0; r < 8; ++r) {
    const int mrow = r + ((lane >> 4) << 3);
    out[mrow * DSZ + n] = acc[r] + bv;
  }
}

// ---------------------------------------------------------------------------
// Kernel 2: norms. blocks 0..127: memory row norms; block 128: query norms
// ---------------------------------------------------------------------------
__global__ __launch_bounds__(256) void k_norms(const float* __restrict__ mem,
                                               float* __restrict__ ws) {
  if (blockIdx.x < 128) {
    const int row = blockIdx.x * 256 + threadIdx.x;
    const v4f* r4 = (const v4f*)(mem + row * DSZ);
    float s = 0.f;
#pragma unroll
    for (int i = 0; i < DSZ / 4; ++i) {
      v4f v = r4[i];
      s += v.x * v.x + v.y * v.y + v.z * v.z + v.w * v.w;
    }
    ws[WS_MN + row] = fmaxf(sqrtf(s), EPSV);
  } else if (threadIdx.x < BATCH) {
    const float* rq = ws + WS_RQ + threadIdx.x * DSZ;
    float s = 0.f;
    for (int i = 0; i < DSZ; ++i) s += rq[i] * rq[i];
    ws[WS_QN + threadIdx.x] = fmaxf(sqrtf(s), EPSV);
  }
}

// ---------------------------------------------------------------------------
// Kernel 3: similarity [16,32768] = (rq @ mem.T)/(qn*mn) via WMMA f32
// one wave per 16x16 tile -> 2048 waves; each memory row read exactly once
// ---------------------------------------------------------------------------
__global__ __launch_bounds__(256) void k_sim(const float* __restrict__ mem,
                                             float* __restrict__ ws) {
  const int wave = (blockIdx.x << 3) + (threadIdx.x >> 5);   // 0..2047
  const int lane = threadIdx.x & 31;
  const int n0   = wave << 4;

  const int m    = lane & 15;
  const int koff = (lane >> 4) << 1;
  const int n    = n0 + (lane & 15);

  const float* arow = ws + WS_RQ + m * DSZ;
  const float* brow = mem + n * DSZ;   // B[k][n] = mem[n][k]

  v8f acc = {};
#pragma unroll 8
  for (int k0 = 0; k0 < DSZ; k0 += 4) {
    v2f a = *(const v2f*)(arow + k0 + koff);
    v2f b = *(const v2f*)(brow + k0 + koff);
    acc = __builtin_amdgcn_wmma_f32_16x16x4_f32(false, a, false, b,
                                                (short)0, acc, false, false);
  }
  const float inv_mn = 1.0f / ws[WS_MN + n];
  float* sim = ws + WS_SIM;
#pragma unroll
  for (int r = 0; r < 8; ++r) {
    const int mrow = r + ((lane >> 4) << 3);
    sim[mrow * MSZ + n] = acc[r] * inv_mn / ws[WS_QN + mrow];
  }
}

// ---------------------------------------------------------------------------
// Kernel 4: per batch row -- top-32 via iterative argmax (register-resident
// row, LDS tree reduce), retrieved gather-sum, output proj, write strength
// ---------------------------------------------------------------------------
__global__ __launch_bounds__(1024) void k_topk(const float* __restrict__ mem,
                                               const float* __restrict__ Wo,
                                               const float* __restrict__ bo,
                                               float* __restrict__ ws,
                                               float* __restrict__ out) {
  __shared__ float red_val[1024];
  __shared__ int   red_idx[1024];
  __shared__ __align__(16) float retr[DSZ];

  const int b = blockIdx.x;
  const int t = threadIdx.x;
  const float* sim = ws + WS_SIM + b * MSZ;

  float v[32];                          // this thread's 32 candidates
#pragma unroll
  for (int i = 0; i < 32; ++i) v[i] = sim[i * 1024 + t];
  if (t < DSZ) retr[t] = 0.f;
  __syncthreads();

  for (int it = 0; it < KTOP; ++it) {
    float mval = v[0];
    int   mi   = 0;
#pragma unroll
    for (int i = 1; i < 32; ++i)
      if (v[i] > mval) { mval = v[i]; mi = i; }
    red_val[t] = mval;
    red_idx[t] = mi * 1024 + t;         // global slot index
    __syncthreads();
    for (int s = 512; s > 0; s >>= 1) {
      if (t < s && red_val[t + s] > red_val[t]) {
        red_val[t] = red_val[t + s];
        red_idx[t] = red_idx[t + s];
      }
      __syncthreads();
    }
    const int jsel = red_idx[0];
    if ((jsel & 1023) == t) {           // owner knocks its element out
      const int li = jsel >> 10;
#pragma unroll
      for (int i = 0; i < 32; ++i)
        if (i == li) v[i] = -__builtin_inff();
    }
    if (t < DSZ) retr[t] += mem[jsel * DSZ + t];
    __syncthreads();
  }

  // output = retrieved @ Wo.T + bo
  float accum = bo[t];
  const v4f* wrow = (const v4f*)(Wo + t * DSZ);
  const v4f* rr   = (const v4f*)retr;
#pragma unroll 8
  for (int i = 0; i < DSZ / 4; ++i) {
    v4f w = wrow[i];
    v4f r = rr[i];
    accum += w.x * r.x + w.y * r.y + w.z * r.z + w.w * r.w;
  }
  out[b * ISZ + t] = accum;

  if (t == 0) {                         // write strength = sigmoid(wv . retr)
    const float* wv = ws + WS_WV + b * DSZ;
    float dot = 0.f;
    for (int i = 0; i < DSZ; ++i) dot += wv[i] * retr[i];
    ws[WS_S + b] = 1.0f / (1.0f + expf(-dot));
  }
}

// ---------------------------------------------------------------------------
// Kernel 5: new_memory[b] = memory + s[b]*wv[b]  (256 MB NT-stored stream)
// ---------------------------------------------------------------------------
__global__ __launch_bounds__(256) void k_update(const float* __restrict__ mem,
                                                const float* __restrict__ ws,
                                                float* __restrict__ out2) {
  const v4f* m4  = (const v4f*)mem;
  const v4f* wv4 = (const v4f*)(ws + WS_WV);
  const float* s = ws + WS_S;
  v4f* o4 = (v4f*)out2;

  const long long total  = (long long)BATCH * MSZ * (DSZ / 4);   // 2^24
  const long long stride = (long long)gridDim.x * blockDim.x;
  for (long long i = (long long)blockIdx.x * blockDim.x + threadIdx.x;
       i < total; i += stride) {
    const int b  = (int)(i >> 20);                  // MSZ*DSZ/4 = 2^20
    const int r  = (int)(i & ((1 << 20) - 1));      // float4 idx in memory
    const int d4 = (int)(i & 31);                   // float4 idx in row
    const float sb = s[b];
    const v4f w  = wv4[b * 32 + d4];
    const v4f mm = m4[r];
    v4f ov;
    ov.x = mm.x + sb * w.x;
    ov.y = mm.y + sb * w.y;
    ov.z = mm.z + sb * w.z;
    ov.w = mm.w + sb * w.w;
    __builtin_nontemporal_store(ov, &o4[i]);
  }
}

// ---------------------------------------------------------------------------
extern "C" void kernel_launch(void* const* d_in, const int* in_sizes, int n_in,
                              void* d_out, int out_size, void* d_ws,
                              size_t ws_size, hipStream_t stream) {
  const float* x   = (const float*)d_in[0];
  const float* mem = (const float*)d_in[1];
  const float* Ww  = (const float*)d_in[2];
  const float* bw  = (const float*)d_in[3];
  const float* Wr  = (const float*)d_in[4];
  const float* br  = (const float*)d_in[5];
  const float* Wo  = (const float*)d_in[6];
  const float* bo  = (const float*)d_in[7];

  float* out  = (float*)d_out;            // [16,1024] then [16,32768,128]
  float* out2 = out + BATCH * ISZ;
  float* ws   = (float*)d_ws;

  k_proj  <<<1,    512, 0, stream>>>(x, Ww, bw, Wr, br, ws);
  k_norms <<<129,  256, 0, stream>>>(mem, ws);
  k_sim   <<<256,  256, 0, stream>>>(mem, ws);
  k_topk  <<<16,  1024, 0, stream>>>(mem, Wo, bo, ws, out);
  k_update<<<4096, 256, 0, stream>>>(mem, ws, out2);
}